// Content_71339406786949
// MI455X (gfx1250) — compile-verified
//
#include <hip/hip_runtime.h>

// Problem constants (from reference): out[b,u] = sum_d x[b,d] * emb[idx[b], d, u]
#define B_N 256
#define D_N 1024
#define U_N 1024
#define C_N 64
// Tiles of 16 examples sharing one content index; worst case:
// <=64 partially-filled group tails + <=256/16 full tiles.
#define MAX_TILES (C_N + B_N / 16) // 80

#define THREADS 256           // 8 waves
#define U_CHUNK 128           // 8 waves x 16-wide u-tiles

typedef __attribute__((ext_vector_type(2))) float v2f;
typedef __attribute__((ext_vector_type(8))) float v8f;

// Workspace (int) layout:
//   ws[0]                      : tile_count
//   ws[1 .. 1+MAX_TILES)       : tile_content[t]
//   ws[1+MAX_TILES .. )        : tile_rows[t][16]  (example index, or -1 = pad)

__global__ void build_tiles_kernel(const int* __restrict__ idx,
                                   int* __restrict__ ws) {
    if (threadIdx.x != 0) return;
    int* tile_content = ws + 1;
    int* tile_rows    = ws + 1 + MAX_TILES;
    int nt = 0;
    for (int c = 0; c < C_N; ++c) {
        int cnt = 0;
        int cur[16];
        for (int b = 0; b < B_N; ++b) {
            if (idx[b] == c) {
                cur[cnt++] = b;
                if (cnt == 16) {
                    tile_content[nt] = c;
                    for (int m = 0; m < 16; ++m) tile_rows[nt * 16 + m] = cur[m];
                    ++nt;
                    cnt = 0;
                }
            }
        }
        if (cnt > 0) {
            tile_content[nt] = c;
            for (int m = 0; m < 16; ++m)
                tile_rows[nt * 16 + m] = (m < cnt) ? cur[m] : -1;
            ++nt;
        }
    }
    ws[0] = nt;
}

// One block = one (tile, 128-wide u-chunk). 8 waves; each wave owns a 16x16
// output tile and iterates K=D in steps of 4 using V_WMMA_F32_16X16X4_F32.
// The weight stream is read once (non-temporal); x is staged in LDS.
__global__ __launch_bounds__(THREADS) void grouped_matvec_wmma(
    const float* __restrict__ x, const float* __restrict__ emb,
    const int* __restrict__ ws, float* __restrict__ out) {
    // Transposed x-tile: xsT[k*16 + m] = x[rows[m]][k].  Exactly 64 KB.
    // A-fragment reads are bank-conflict-free: lanes 0-15 -> banks l (+0/+16),
    // lanes 16-31 -> banks l+32 (+0/+16).
    __shared__ float xsT[D_N * 16];

    const int t = blockIdx.y;
    if (t >= ws[0]) return; // uniform per block: EXEC stays all-ones for WMMA

    const int c = ws[1 + t];
    const int* __restrict__ rows = ws + 1 + MAX_TILES + t * 16;

    const int tid = threadIdx.x;

    // Stage gathered x rows into LDS (conflict-free ds writes: consecutive
    // lanes write consecutive LDS addresses). x is small & reused -> cached.
    for (int i = tid; i < 16 * D_N; i += THREADS) {
        const int m = i & 15;
        const int k = i >> 4;
        const int r = rows[m];
        const float v = (r >= 0) ? x[(size_t)r * D_N + k] : 0.0f;
        xsT[k * 16 + m] = v;
    }
    __syncthreads();

    const int wave = tid >> 5;
    const int lane = tid & 31;
    const int half = lane >> 4;  // K-pair selector for A/B fragments
    const int l16  = lane & 15;  // M index (A) / N index (B,C,D)
    const int u0   = blockIdx.x * U_CHUNK + wave * 16;

    // B-fragment base: W[c][d][u0 + l16]; lanes 0-15 cover 64B-contiguous u.
    const float* __restrict__ wbase =
        emb + (size_t)c * (D_N * U_N) + (size_t)u0 + l16;

    v8f acc = {};
#pragma unroll 8
    for (int k0 = 0; k0 < D_N; k0 += 4) {
        const int ka = k0 + 2 * half;
        // A 16x4 fragment (ISA layout): lane l -> M=l&15, VGPR j -> K=2*(l>>4)+j
        v2f a;
        a.x = xsT[ka * 16 + l16];
        a.y = xsT[(ka + 1) * 16 + l16];
        // B 4x16 fragment (mirrored layout): lane l -> N=l&15, VGPR j -> K=2*(l>>4)+j
        // Streamed exactly once -> non-temporal (TH=NT): keep L2 for x/out.
        v2f b;
        const float* __restrict__ wp = wbase + (size_t)ka * U_N;
        b.x = __builtin_nontemporal_load(wp);
        b.y = __builtin_nontemporal_load(wp + U_N);
        // D = A*B + C, full fp32 (matches reference precision)
        acc = __builtin_amdgcn_wmma_f32_16x16x4_f32(
            /*neg_a=*/false, a, /*neg_b=*/false, b,
            /*c_mod=*/(short)0, acc, /*reuse_a=*/false, /*reuse_b=*/false);
    }

    // C/D fragment layout: lane l -> N=l&15; VGPR v -> M = v + 8*(l>>4).
#pragma unroll
    for (int v = 0; v < 8; ++v) {
        const int m = v + 8 * half;
        const int r = rows[m];
        if (r >= 0)
            __builtin_nontemporal_store(acc[v],
                                        out + (size_t)r * U_N + u0 + l16);
    }
}

extern "C" void kernel_launch(void* const* d_in, const int* in_sizes, int n_in,
                              void* d_out, int out_size, void* d_ws,
                              size_t ws_size, hipStream_t stream) {
    const int*   idx = (const int*)d_in[0];
    const float* x   = (const float*)d_in[1];
    const float* emb = (const float*)d_in[2];
    float*       out = (float*)d_out;
    int*         ws  = (int*)d_ws;

    build_tiles_kernel<<<1, 32, 0, stream>>>(idx, ws);

    dim3 grid(U_N / U_CHUNK, MAX_TILES);
    grouped_matvec_wmma<<<grid, THREADS, 0, stream>>>(x, emb, ws, out);
}